// DeformBottleneckBlock_19464791785705
// MI455X (gfx1250) — compile-verified
//
#include <hip/hip_runtime.h>
#include <math.h>

// ---------------------------------------------------------------------------
// DeformBottleneckBlock on gfx1250 (MI455X), fp32, WMMA f32 16x16x4 path.
// B=4, C_in=1024, C_mid=256, H=W=48 (HW=2304 = 144 * 16 tiles).
// Register-blocked: dense GEMMs 64x32 per wave (8 WMMA accums), deform GEMM
// 64x16 per wave (gathered B fragment reused across 4 WMMAs).
// ---------------------------------------------------------------------------

typedef float v2f __attribute__((ext_vector_type(2)));
typedef float v8f __attribute__((ext_vector_type(8)));

#define BATCH  4
#define HH     48
#define WW     48
#define HW     2304
#define C_IN   1024
#define C_MID  256
#define C_OUT  1024
#define EPSV   1e-5f

static __device__ __forceinline__ v8f wmma_f32(v2f a, v2f b, v8f c) {
  // (neg_a, A, neg_b, B, c_mod, C, reuse_a, reuse_b)
  return __builtin_amdgcn_wmma_f32_16x16x4_f32(false, a, false, b, (short)0, c,
                                               false, false);
}

// ---------------------------------------------------------------------------
// Kernel A: 1x1 conv 1024->256 as GEMM, fused BN1 + ReLU.
// One wave per 64(out-ch) x 32(pixel) macro-tile; 4 waves / block.
// A-frag (W, 16x4): lane L row = m0+16*mi+(L&15), K pair = k + 2*(L>>4).
// B-frag (X, 4x16): lane L col = n0+16*ni+(L&15), K pair = k + 2*(L>>4).
// ---------------------------------------------------------------------------
__global__ __launch_bounds__(128) void
conv1_bn_relu_kernel(const float* __restrict__ x, const float* __restrict__ w1,
                     const float* __restrict__ g, const float* __restrict__ bt,
                     const float* __restrict__ mn, const float* __restrict__ vr,
                     float* __restrict__ out1) {
  const int lane = threadIdx.x & 31;
  int tile = blockIdx.x * 4 + (threadIdx.x >> 5);   // 4 * 4 * 72 = 1152 tiles
  const int nt = tile % (HW / 32); tile /= (HW / 32);
  const int mt = tile % (C_MID / 64);
  const int b  = tile / (C_MID / 64);

  const int p0  = nt * 32 + (lane & 15);
  const int r0  = mt * 64 + (lane & 15);
  const int khi = (lane >> 4) << 1;

  const float* __restrict__ xb = x + (size_t)b * C_IN * HW;
  const float* wr[4];
  #pragma unroll
  for (int mi = 0; mi < 4; ++mi) wr[mi] = w1 + (size_t)(r0 + 16 * mi) * C_IN;

  v8f acc[4][2];
  #pragma unroll
  for (int mi = 0; mi < 4; ++mi)
    #pragma unroll
    for (int ni = 0; ni < 2; ++ni)
      acc[mi][ni] = (v8f){0.f, 0.f, 0.f, 0.f, 0.f, 0.f, 0.f, 0.f};

  for (int k = 0; k < C_IN; k += 4) {
    const int kk = k + khi;
    const float* __restrict__ b0 = xb + (size_t)kk * HW;
    const float* __restrict__ b1 = b0 + HW;
    if (k + 8 < C_IN) {                       // hide L2->L1 latency (uniform)
      __builtin_prefetch(b0 + 8 * HW + p0, 0, 3);
      __builtin_prefetch(b1 + 8 * HW + p0, 0, 3);
    }
    v2f bv0, bv1;
    bv0.x = b0[p0];      bv0.y = b1[p0];
    bv1.x = b0[p0 + 16]; bv1.y = b1[p0 + 16];
    #pragma unroll
    for (int mi = 0; mi < 4; ++mi) {
      v2f a = *reinterpret_cast<const v2f*>(wr[mi] + kk);   // 8B aligned
      acc[mi][0] = wmma_f32(a, bv0, acc[mi][0]);
      acc[mi][1] = wmma_f32(a, bv1, acc[mi][1]);
    }
  }

  const int mb = mt * 64 + ((lane >> 4) << 3);
  float* __restrict__ ob = out1 + (size_t)b * C_MID * HW;
  #pragma unroll
  for (int mi = 0; mi < 4; ++mi) {
    #pragma unroll
    for (int r = 0; r < 8; ++r) {
      const int mm = mb + 16 * mi + r;
      const float inv  = g[mm] * rsqrtf(vr[mm] + EPSV);
      const float bias = bt[mm] - mn[mm] * inv;
      ob[(size_t)mm * HW + p0]      = fmaxf(acc[mi][0][r] * inv + bias, 0.f);
      ob[(size_t)mm * HW + p0 + 16] = fmaxf(acc[mi][1][r] * inv + bias, 0.f);
    }
  }
}

// ---------------------------------------------------------------------------
// Kernel B: 3x3 offset conv 256->27, pad 1, + bias. Tiny (1.1 GFLOP): plain VALU.
// ---------------------------------------------------------------------------
__global__ __launch_bounds__(256) void
offset_conv_kernel(const float* __restrict__ feat, const float* __restrict__ ow,
                   const float* __restrict__ obias, float* __restrict__ om) {
  int t = blockIdx.x * blockDim.x + threadIdx.x;
  if (t >= BATCH * 27 * HW) return;
  const int p  = t % HW;
  int tmp      = t / HW;
  const int oc = tmp % 27;
  const int b  = tmp / 27;
  const int y  = p / WW;
  const int xp = p % WW;

  const float* __restrict__ fb = feat + (size_t)b * C_MID * HW;
  float acc = obias[oc];
  #pragma unroll
  for (int ky = 0; ky < 3; ++ky) {
    const int iy = y + ky - 1;
    if (iy < 0 || iy >= HH) continue;
    #pragma unroll
    for (int kx = 0; kx < 3; ++kx) {
      const int ix = xp + kx - 1;
      if (ix < 0 || ix >= WW) continue;
      const float* __restrict__ f = fb + iy * WW + ix;
      const float* __restrict__ w = ow + (size_t)oc * C_MID * 9 + ky * 3 + kx;
      for (int ic = 0; ic < C_MID; ++ic)
        acc = fmaf(f[(size_t)ic * HW], w[(size_t)ic * 9], acc);
    }
  }
  om[(size_t)b * 27 * HW + (size_t)oc * HW + p] = acc;
}

// ---------------------------------------------------------------------------
// Kernel C: precompute bilinear corner indices + mask-folded weights per
// (b, tap k, pixel). Amortized over 256 channels in Kernel D.
// ---------------------------------------------------------------------------
__global__ __launch_bounds__(256) void
build_table_kernel(const float* __restrict__ om, int4* __restrict__ tidx,
                   float4* __restrict__ tw) {
  int t = blockIdx.x * blockDim.x + threadIdx.x;
  if (t >= BATCH * 9 * HW) return;
  const int p = t % HW;
  int tmp     = t / HW;
  const int k = tmp % 9;
  const int b = tmp / 9;
  const int y  = p / WW;
  const int xp = p % WW;

  const float* __restrict__ omb = om + (size_t)b * 27 * HW;
  const float ox = omb[(size_t)k * HW + p];
  const float oy = omb[(size_t)(9 + k) * HW + p];
  float msk      = omb[(size_t)(18 + k) * HW + p];
  msk = 1.f / (1.f + __expf(-msk));                  // sigmoid

  const float py = (float)y  + (float)(k / 3 - 1) + oy;
  const float px = (float)xp + (float)(k % 3 - 1) + ox;
  const float y0f = floorf(py), x0f = floorf(px);
  const float wy1 = py - y0f,  wx1 = px - x0f;
  const int y0 = (int)y0f, x0 = (int)x0f, y1 = y0 + 1, x1 = x0 + 1;

  const float vy0 = (y0 >= 0 && y0 < HH) ? 1.f : 0.f;
  const float vy1 = (y1 >= 0 && y1 < HH) ? 1.f : 0.f;
  const float vx0 = (x0 >= 0 && x0 < WW) ? 1.f : 0.f;
  const float vx1 = (x1 >= 0 && x1 < WW) ? 1.f : 0.f;
  const int cy0 = min(max(y0, 0), HH - 1), cy1 = min(max(y1, 0), HH - 1);
  const int cx0 = min(max(x0, 0), WW - 1), cx1 = min(max(x1, 0), WW - 1);

  int4 idx;
  idx.x = cy0 * WW + cx0; idx.y = cy0 * WW + cx1;
  idx.z = cy1 * WW + cx0; idx.w = cy1 * WW + cx1;
  float4 wv;
  wv.x = (1.f - wy1) * (1.f - wx1) * vy0 * vx0 * msk;
  wv.y = (1.f - wy1) * wx1         * vy0 * vx1 * msk;
  wv.z = wy1         * (1.f - wx1) * vy1 * vx0 * msk;
  wv.w = wy1         * wx1         * vy1 * vx1 * msk;
  tidx[t] = idx;
  tw[t]   = wv;
}

// ---------------------------------------------------------------------------
// Kernel D: deformable 3x3 conv as GEMM (M=256, K=9*256, N=HW).
// 64(M) x 16(N) per wave: each gathered B fragment feeds 4 WMMAs.
// Fused BN2 + ReLU.
// ---------------------------------------------------------------------------
__global__ __launch_bounds__(128) void
deform_bn_relu_kernel(const float* __restrict__ feat, const float* __restrict__ w2,
                      const int4* __restrict__ tidx, const float4* __restrict__ tw,
                      const float* __restrict__ g, const float* __restrict__ bt,
                      const float* __restrict__ mn, const float* __restrict__ vr,
                      float* __restrict__ out2) {
  const int lane = threadIdx.x & 31;
  int tile = blockIdx.x * 4 + (threadIdx.x >> 5);   // 4 * 4 * 144 = 2304 tiles
  const int nt = tile % (HW / 16); tile /= (HW / 16);
  const int mt = tile % (C_MID / 64);
  const int b  = tile / (C_MID / 64);

  const int p   = nt * 16 + (lane & 15);
  const int r0  = mt * 64 + (lane & 15);
  const int khi = (lane >> 4) << 1;

  const float* __restrict__ fb = feat + (size_t)b * C_MID * HW;

  v8f acc[4];
  #pragma unroll
  for (int mi = 0; mi < 4; ++mi)
    acc[mi] = (v8f){0.f, 0.f, 0.f, 0.f, 0.f, 0.f, 0.f, 0.f};

  for (int k = 0; k < 9; ++k) {
    const int te    = (b * 9 + k) * HW + p;
    const int4  idx = tidx[te];
    const float4 wv = tw[te];
    // w2 layout [oc][c][9]
    const float* wk[4];
    #pragma unroll
    for (int mi = 0; mi < 4; ++mi)
      wk[mi] = w2 + (size_t)(r0 + 16 * mi) * C_MID * 9 + k;

    for (int c = 0; c < C_MID; c += 4) {
      const int kk = c + khi;
      const float* __restrict__ f0 = fb + (size_t)kk * HW;
      const float* __restrict__ f1 = f0 + HW;
      v2f bv;
      bv.x = wv.x * f0[idx.x] + wv.y * f0[idx.y] + wv.z * f0[idx.z] + wv.w * f0[idx.w];
      bv.y = wv.x * f1[idx.x] + wv.y * f1[idx.y] + wv.z * f1[idx.z] + wv.w * f1[idx.w];
      #pragma unroll
      for (int mi = 0; mi < 4; ++mi) {
        v2f a;
        a.x = wk[mi][(size_t)kk * 9];
        a.y = wk[mi][(size_t)(kk + 1) * 9];
        acc[mi] = wmma_f32(a, bv, acc[mi]);
      }
    }
  }

  const int mb = mt * 64 + ((lane >> 4) << 3);
  float* __restrict__ ob = out2 + (size_t)b * C_MID * HW;
  #pragma unroll
  for (int mi = 0; mi < 4; ++mi) {
    #pragma unroll
    for (int r = 0; r < 8; ++r) {
      const int mm = mb + 16 * mi + r;
      const float inv = g[mm] * rsqrtf(vr[mm] + EPSV);
      const float val = acc[mi][r] * inv + (bt[mm] - mn[mm] * inv);
      ob[(size_t)mm * HW + p] = fmaxf(val, 0.f);
    }
  }
}

// ---------------------------------------------------------------------------
// Kernel E: 1x1 conv 256->1024 as GEMM, fused BN3 + residual(x) + ReLU.
// 64(M) x 32(N) per wave, same blocking as Kernel A.
// ---------------------------------------------------------------------------
__global__ __launch_bounds__(128) void
conv3_bn_add_relu_kernel(const float* __restrict__ in2, const float* __restrict__ w3,
                         const float* __restrict__ g, const float* __restrict__ bt,
                         const float* __restrict__ mn, const float* __restrict__ vr,
                         const float* __restrict__ xres, float* __restrict__ out) {
  const int lane = threadIdx.x & 31;
  int tile = blockIdx.x * 4 + (threadIdx.x >> 5);   // 4 * 16 * 72 = 4608 tiles
  const int nt = tile % (HW / 32); tile /= (HW / 32);
  const int mt = tile % (C_OUT / 64);
  const int b  = tile / (C_OUT / 64);

  const int p0  = nt * 32 + (lane & 15);
  const int r0  = mt * 64 + (lane & 15);
  const int khi = (lane >> 4) << 1;

  const float* __restrict__ ib = in2 + (size_t)b * C_MID * HW;
  const float* wr[4];
  #pragma unroll
  for (int mi = 0; mi < 4; ++mi) wr[mi] = w3 + (size_t)(r0 + 16 * mi) * C_MID;

  v8f acc[4][2];
  #pragma unroll
  for (int mi = 0; mi < 4; ++mi)
    #pragma unroll
    for (int ni = 0; ni < 2; ++ni)
      acc[mi][ni] = (v8f){0.f, 0.f, 0.f, 0.f, 0.f, 0.f, 0.f, 0.f};

  for (int k = 0; k < C_MID; k += 4) {
    const int kk = k + khi;
    const float* __restrict__ b0 = ib + (size_t)kk * HW;
    const float* __restrict__ b1 = b0 + HW;
    if (k + 8 < C_MID) {
      __builtin_prefetch(b0 + 8 * HW + p0, 0, 3);
      __builtin_prefetch(b1 + 8 * HW + p0, 0, 3);
    }
    v2f bv0, bv1;
    bv0.x = b0[p0];      bv0.y = b1[p0];
    bv1.x = b0[p0 + 16]; bv1.y = b1[p0 + 16];
    #pragma unroll
    for (int mi = 0; mi < 4; ++mi) {
      v2f a = *reinterpret_cast<const v2f*>(wr[mi] + kk);
      acc[mi][0] = wmma_f32(a, bv0, acc[mi][0]);
      acc[mi][1] = wmma_f32(a, bv1, acc[mi][1]);
    }
  }

  const int mb = mt * 64 + ((lane >> 4) << 3);
  const float* __restrict__ xb = xres + (size_t)b * C_OUT * HW;
  float* __restrict__ ob       = out  + (size_t)b * C_OUT * HW;
  #pragma unroll
  for (int mi = 0; mi < 4; ++mi) {
    #pragma unroll
    for (int r = 0; r < 8; ++r) {
      const int mm = mb + 16 * mi + r;
      const float inv  = g[mm] * rsqrtf(vr[mm] + EPSV);
      const float bias = bt[mm] - mn[mm] * inv;
      const float v0 = acc[mi][0][r] * inv + bias + xb[(size_t)mm * HW + p0];
      const float v1 = acc[mi][1][r] * inv + bias + xb[(size_t)mm * HW + p0 + 16];
      ob[(size_t)mm * HW + p0]      = fmaxf(v0, 0.f);
      ob[(size_t)mm * HW + p0 + 16] = fmaxf(v1, 0.f);
    }
  }
}

// ---------------------------------------------------------------------------
// Workspace layout (bytes), all 16B aligned:
//   out1: [0,            9437184)   4*256*2304 f32
//   out2: [9437184,     18874368)
//   om  : [18874368,    19869696)   4*27*2304 f32
//   tidx: [19869696,    21196800)   4*9*2304 int4
//   tw  : [21196800,    22523904)   4*9*2304 float4
// ---------------------------------------------------------------------------
extern "C" void kernel_launch(void* const* d_in, const int* in_sizes, int n_in,
                              void* d_out, int out_size, void* d_ws, size_t ws_size,
                              hipStream_t stream) {
  (void)in_sizes; (void)n_in; (void)out_size; (void)ws_size;

  const float* x      = (const float*)d_in[0];
  const float* w1     = (const float*)d_in[1];
  const float* bn1_g  = (const float*)d_in[2];
  const float* bn1_b  = (const float*)d_in[3];
  const float* bn1_m  = (const float*)d_in[4];
  const float* bn1_v  = (const float*)d_in[5];
  const float* off_w  = (const float*)d_in[6];
  const float* off_b  = (const float*)d_in[7];
  const float* w2     = (const float*)d_in[8];
  const float* bn2_g  = (const float*)d_in[9];
  const float* bn2_b  = (const float*)d_in[10];
  const float* bn2_m  = (const float*)d_in[11];
  const float* bn2_v  = (const float*)d_in[12];
  const float* w3     = (const float*)d_in[13];
  const float* bn3_g  = (const float*)d_in[14];
  const float* bn3_b  = (const float*)d_in[15];
  const float* bn3_m  = (const float*)d_in[16];
  const float* bn3_v  = (const float*)d_in[17];

  char* ws = (char*)d_ws;
  float*  out1 = (float*)(ws + 0);
  float*  out2 = (float*)(ws + 9437184);
  float*  om   = (float*)(ws + 18874368);
  int4*   tidx = (int4*)(ws + 19869696);
  float4* tw   = (float4*)(ws + 21196800);

  // A: conv1 + BN1 + ReLU   (1152 wave-tiles of 64x32, 4 waves/block)
  conv1_bn_relu_kernel<<<288, 128, 0, stream>>>(x, w1, bn1_g, bn1_b, bn1_m,
                                                bn1_v, out1);
  // B: offset conv (27 ch)
  offset_conv_kernel<<<(BATCH * 27 * HW + 255) / 256, 256, 0, stream>>>(
      out1, off_w, off_b, om);
  // C: bilinear corner/weight table
  build_table_kernel<<<(BATCH * 9 * HW + 255) / 256, 256, 0, stream>>>(
      om, tidx, tw);
  // D: deformable conv + BN2 + ReLU (2304 wave-tiles of 64x16)
  deform_bn_relu_kernel<<<576, 128, 0, stream>>>(out1, w2, tidx, tw, bn2_g,
                                                 bn2_b, bn2_m, bn2_v, out2);
  // E: conv3 + BN3 + residual + ReLU (4608 wave-tiles of 64x32)
  conv3_bn_add_relu_kernel<<<1152, 128, 0, stream>>>(out2, w3, bn3_g, bn3_b,
                                                     bn3_m, bn3_v, x,
                                                     (float*)d_out);
}